// ProtoNet_6966436954815
// MI455X (gfx1250) — compile-verified
//
#include <hip/hip_runtime.h>
#include <math.h>

typedef float v2f __attribute__((ext_vector_type(2)));
typedef float v8f __attribute__((ext_vector_type(8)));

#define D      640
#define NWAY   5
#define NSHOT  5
#define NQTOT  75
#define TOPK   16
#define MAXBLK 512
#define WPB    8

#define APAIR_OFF  0        // (D/2)*32 = 10240 floats
#define SIMSUP_OFF 10240    // 25 floats
#define BV_OFF     10272    // nblk*NWAY*TOPK floats, then same count of ints

__device__ inline float wave_sum(float v) {
  v += __shfl_xor(v, 16, 32);
  v += __shfl_xor(v, 8, 32);
  v += __shfl_xor(v, 4, 32);
  v += __shfl_xor(v, 2, 32);
  v += __shfl_xor(v, 1, 32);
  return v;
}

__device__ inline bool better(float v, int id, float tv, int ti) {
  return (v > tv) || (v == tv && id < ti);
}

__device__ inline void insert16(float v, int id, float* tv, int* ti) {
  bool b[TOPK];
#pragma unroll
  for (int j = 0; j < TOPK; ++j) b[j] = better(v, id, tv[j], ti[j]);
#pragma unroll
  for (int j = TOPK - 1; j >= 1; --j) {
    if (b[j]) { tv[j] = b[j - 1] ? tv[j - 1] : v; ti[j] = b[j - 1] ? ti[j - 1] : id; }
  }
  if (b[0]) { tv[0] = v; ti[0] = id; }
}

// ---------------- Kernel A: support prep -----------------------------------
__global__ void ka_prep(const float* __restrict__ support, float* __restrict__ ws) {
  __shared__ float s_invn[NWAY * NSHOT];
  __shared__ float s_a[NWAY * D];
  int tid = threadIdx.x;
  int lane = tid & 31, wv = tid >> 5;

  // phase 1: inverse norms of each (way, shot) support vector
  for (int p = wv; p < NWAY * NSHOT; p += WPB) {
    int w = p / NSHOT, s = p % NSHOT;
    const float* x = support + (size_t)(s * NWAY + w) * D;
    float acc = 0.f;
    for (int k = lane; k < D; k += 32) { float t = x[k]; acc += t * t; }
    acc = wave_sum(acc);
    if (lane == 0) s_invn[p] = 1.0f / fmaxf(sqrtf(acc), 1e-12f);
  }
  __syncthreads();

  // phase 2: a[w][k] = mean_s sup_n[w,s,k]
  for (int i = tid; i < NWAY * D; i += blockDim.x) {
    int w = i / D, k = i % D;
    float acc = 0.f;
#pragma unroll
    for (int s = 0; s < NSHOT; ++s)
      acc += support[(size_t)(s * NWAY + w) * D + k] * s_invn[w * NSHOT + s];
    s_a[i] = acc * (1.0f / (float)NSHOT);
  }
  __syncthreads();

  // phase 2b: write B fragments: aPair[(k/2)*32 + n*2 + {0,1}] = aT[k][n], aT[k+1][n]
  for (int i = tid; i < (D / 2) * 32; i += blockDim.x) {
    int kp = i >> 5, r = i & 31, n = r >> 1, k = (kp << 1) | (r & 1);
    ws[APAIR_OFF + i] = (n < NWAY) ? s_a[n * D + k] : 0.0f;
  }

  // phase 3: sim_sup[w][t] = a[w] . sup_n[w,t]
  for (int p = wv; p < NWAY * NSHOT; p += WPB) {
    int w = p / NSHOT, t = p % NSHOT;
    const float* x = support + (size_t)(t * NWAY + w) * D;
    float acc = 0.f;
    for (int k = lane; k < D; k += 32) acc += s_a[w * D + k] * x[k];
    acc = wave_sum(acc);
    if (lane == 0) ws[SIMSUP_OFF + p] = acc * s_invn[w * NSHOT + t];
  }
}

// ---------------- Kernel B: streaming memory scan with WMMA ----------------
__global__ void kb_scan(const float* __restrict__ memory,
                        const float* __restrict__ aPair,
                        float* __restrict__ bv, int* __restrict__ bi, int n_mem) {
  __shared__ __align__(16) float tileA[WPB * 16 * 68];
  __shared__ float mgV[NWAY][2 * WPB * TOPK];
  __shared__ int   mgI[NWAY][2 * WPB * TOPK];
  int lane = threadIdx.x & 31, wid = threadIdx.x >> 5;
  int half = lane >> 4, lrow = lane & 15;
  float* tw = tileA + wid * (16 * 68);

  float topV[TOPK]; int topI[TOPK];
#pragma unroll
  for (int j = 0; j < TOPK; ++j) { topV[j] = -__builtin_inff(); topI[j] = 0x7fffffff; }

  int ntile = (n_mem + 15) >> 4;
  int stride = gridDim.x * WPB;
  for (int tile = blockIdx.x * WPB + wid; tile < ntile; tile += stride) {
    int base = tile << 4;
    v8f c = {0.f, 0.f, 0.f, 0.f, 0.f, 0.f, 0.f, 0.f};
    float n2p = 0.f;
    for (int c0 = 0; c0 < D; c0 += 64) {
      // stage 16 rows x 64 cols into LDS (coalesced 256B half-wave segments)
#pragma unroll
      for (int it = 0; it < 8; ++it) {
        int r = it * 2 + half;
        int col = lrow * 4;
        int row = base + r;
        float4 v = make_float4(0.f, 0.f, 0.f, 0.f);
        if (row < n_mem)
          v = *(const float4*)(memory + (size_t)row * D + c0 + col);
        *(float4*)(tw + r * 68 + col) = v;
      }
      __builtin_amdgcn_wave_barrier();
      // 16 WMMA steps: D[row][way] += mem[row][k..k+3] * aT[k..k+3][way]
#pragma unroll
      for (int s = 0; s < 16; ++s) {
        v2f a = *(const v2f*)(tw + lrow * 68 + 4 * s + 2 * half);
        int kb = c0 + 4 * s + 2 * half;
        v2f b = *(const v2f*)(aPair + (kb >> 1) * 32 + lrow * 2);
        c = __builtin_amdgcn_wmma_f32_16x16x4_f32(false, a, false, b, (short)0, c,
                                                  false, false);
      }
      // per-row norm partials (lane handles row lrow, half of the 64 cols)
#pragma unroll
      for (int j = 0; j < 8; ++j) {
        float4 t = *(const float4*)(tw + lrow * 68 + half * 32 + j * 4);
        n2p += t.x * t.x + t.y * t.y + t.z * t.z + t.w * t.w;
      }
      __builtin_amdgcn_wave_barrier();
    }
    float n2 = n2p + __shfl_xor(n2p, 16, 32);  // full ||row lrow||^2 on every lane
#pragma unroll
    for (int i = 0; i < 8; ++i) {
      int r = i + 8 * half;
      float nr = __shfl(n2, r, 32);
      int row = base + r;
      float simv = 0.2f * c[i] / fmaxf(sqrtf(nr), 1e-12f);
      int cid = row + NSHOT;  // index in concat(sim_sup, sim_mem)
      if (row < n_mem && better(simv, cid, topV[TOPK - 1], topI[TOPK - 1]))
        insert16(simv, cid, topV, topI);
    }
  }

  // block-level merge: each wave dumps per-way lists, wave 0 merges
  if (lrow < NWAY) {
    int slot = (wid * 2 + half) * TOPK;
#pragma unroll
    for (int j = 0; j < TOPK; ++j) { mgV[lrow][slot + j] = topV[j]; mgI[lrow][slot + j] = topI[j]; }
  }
  __syncthreads();
  if (wid == 0 && lane < NWAY) {
    float fv[TOPK]; int fi[TOPK];
#pragma unroll
    for (int j = 0; j < TOPK; ++j) { fv[j] = -__builtin_inff(); fi[j] = 0x7fffffff; }
    for (int e = 0; e < 2 * WPB * TOPK; ++e) {
      float v = mgV[lane][e]; int id = mgI[lane][e];
      if (better(v, id, fv[TOPK - 1], fi[TOPK - 1])) insert16(v, id, fv, fi);
    }
#pragma unroll
    for (int j = 0; j < TOPK; ++j) {
      bv[((size_t)blockIdx.x * NWAY + lane) * TOPK + j] = fv[j];
      bi[((size_t)blockIdx.x * NWAY + lane) * TOPK + j] = fi[j];
    }
  }
}

// ---------------- Kernel C: final merge, prototypes, logits ----------------
__global__ void kc_final(const float* __restrict__ support, const float* __restrict__ query,
                         const float* __restrict__ memory, const float* __restrict__ ws,
                         const float* __restrict__ bv, const int* __restrict__ bi,
                         float* __restrict__ out, int nblk) {
  __shared__ float lv[NWAY][32 * TOPK];
  __shared__ int   li[NWAY][32 * TOPK];
  __shared__ float finV[NWAY][TOPK];
  __shared__ int   finI[NWAY][TOPK];
  __shared__ float proto[NWAY * D];
  __shared__ float invp[NWAY];
  int lane = threadIdx.x & 31, wv = threadIdx.x >> 5;

  // per-lane partial top-16 over block candidates
  if (wv < NWAY) {
    int way = wv;
    float tv[TOPK]; int ti[TOPK];
#pragma unroll
    for (int j = 0; j < TOPK; ++j) { tv[j] = -__builtin_inff(); ti[j] = 0x7fffffff; }
    int total = nblk * TOPK;
    for (int e = lane; e < total; e += 32) {
      int blk = e / TOPK, j = e % TOPK;
      float v = bv[((size_t)blk * NWAY + way) * TOPK + j];
      int id = bi[((size_t)blk * NWAY + way) * TOPK + j];
      if (better(v, id, tv[TOPK - 1], ti[TOPK - 1])) insert16(v, id, tv, ti);
    }
#pragma unroll
    for (int j = 0; j < TOPK; ++j) { lv[way][lane * TOPK + j] = tv[j]; li[way][lane * TOPK + j] = ti[j]; }
  }
  __syncthreads();
  if (wv < NWAY && lane == 0) {
    int way = wv;
    float tv[TOPK]; int ti[TOPK];
#pragma unroll
    for (int j = 0; j < TOPK; ++j) { tv[j] = -__builtin_inff(); ti[j] = 0x7fffffff; }
    for (int t = 0; t < NSHOT; ++t) {  // support entries (concat indices 0..4)
      float v = ws[SIMSUP_OFF + way * NSHOT + t];
      if (better(v, t, tv[TOPK - 1], ti[TOPK - 1])) insert16(v, t, tv, ti);
    }
    for (int e = 0; e < 32 * TOPK; ++e) {
      float v = lv[way][e]; int id = li[way][e];
      if (better(v, id, tv[TOPK - 1], ti[TOPK - 1])) insert16(v, id, tv, ti);
    }
#pragma unroll
    for (int j = 0; j < TOPK; ++j) { finV[way][j] = tv[j]; finI[way][j] = ti[j]; }
  }
  __syncthreads();

  // prototypes: weighted sum of gathered embeddings, then l2-normalize
  if (wv < NWAY) {
    int way = wv;
    float wsumv = 0.f;
#pragma unroll
    for (int j = 0; j < TOPK; ++j) wsumv += finV[way][j];
    float inv_ws = 1.0f / wsumv;
    float np = 0.f;
    for (int k = lane; k < D; k += 32) {
      float acc = 0.f;
#pragma unroll
      for (int j = 0; j < TOPK; ++j) {
        int id = finI[way][j];
        float e = (id < NSHOT) ? support[(size_t)(id * NWAY + way) * D + k]
                               : memory[(size_t)(id - NSHOT) * D + k];
        acc += finV[way][j] * e;
      }
      float p = acc * inv_ws;
      proto[way * D + k] = p;
      np += p * p;
    }
    np = wave_sum(np);
    if (lane == 0) invp[way] = 1.0f / fmaxf(sqrtf(np), 1e-12f);
  }
  __syncthreads();

  // logits[q][w] = (query[q] . proto_n[w]) / 64
  for (int q = wv; q < NQTOT; q += WPB) {
    for (int w = 0; w < NWAY; ++w) {
      float acc = 0.f;
      for (int k = lane; k < D; k += 32) acc += query[(size_t)q * D + k] * proto[w * D + k];
      acc = wave_sum(acc);
      if (lane == 0) out[q * NWAY + w] = acc * invp[w] * (1.0f / 64.0f);
    }
  }
}

extern "C" void kernel_launch(void* const* d_in, const int* in_sizes, int n_in,
                              void* d_out, int out_size, void* d_ws, size_t ws_size,
                              hipStream_t stream) {
  const float* support = (const float*)d_in[0];
  const float* query   = (const float*)d_in[1];
  const float* memory  = (const float*)d_in[2];
  float* out = (float*)d_out;
  float* ws  = (float*)d_ws;
  int n_mem = in_sizes[2] / D;

  // Scale block count to available scratch: each block needs NWAY*TOPK floats
  // + NWAY*TOPK ints of candidate storage. More blocks = more waves in flight
  // to saturate 23.3 TB/s; cap at 512 (4096 waves for 18750 tiles).
  long avail = (long)(ws_size / 4) - BV_OFF;
  long per_blk = 2L * NWAY * TOPK;
  int nblk = (int)(avail / per_blk);
  if (nblk > MAXBLK) nblk = MAXBLK;
  if (nblk < 1) nblk = 1;

  float* bv = ws + BV_OFF;
  int*   bi = (int*)(ws + BV_OFF + (size_t)nblk * NWAY * TOPK);

  ka_prep<<<1, 256, 0, stream>>>(support, ws);
  kb_scan<<<nblk, 256, 0, stream>>>(memory, ws + APAIR_OFF, bv, bi, n_mem);
  kc_final<<<1, 256, 0, stream>>>(support, query, memory, ws, bv, bi, out, nblk);
}